// CTCLayer_42288247996866
// MI455X (gfx1250) — compile-verified
//
#include <hip/hip_runtime.h>
#include <hip/hip_bf16.h>
#include <math.h>

typedef __attribute__((ext_vector_type(2))) float v2f;
typedef __attribute__((ext_vector_type(8))) float v8f;

#define CTC_B     1024
#define CTC_T     256
#define CTC_C     96
#define CTC_L     32
#define CTC_BLANK 95
#define NEGV      (-1e30f)
#define EPSV      (1e-7f)

// ---------------------------------------------------------------------------
// Kernel A: per-(b,t) row sums of y_pred over C=96 classes, via
// V_WMMA_F32_16X16X4_F32 with a ones B-matrix. Each wave reduces 16 rows:
//   A tile: 16x4 f32 chunk of rows (lane<16 -> K=0,1 ; lane>=16 -> K=2,3)
//   B     : 4x16 all-ones  ->  D[m][n] = rowsum(m), replicated over n.
// 24 accumulator-chained WMMAs cover K=96. Output: nrm[row]=log(sum + C*eps),
// i.e. the log-softmax denominator (exp cancels: logsumexp(log(p+eps))).
// ---------------------------------------------------------------------------
__global__ void ctc_rowsum_wmma(const float* __restrict__ P,
                                float* __restrict__ nrm, int nrows) {
  const int lane = threadIdx.x & 31;
  const int wave = blockIdx.x * (blockDim.x >> 5) + (threadIdx.x >> 5);
  const int r0 = wave * 16;
  if (r0 >= nrows) return;                 // wave-uniform: EXEC stays all-1s

  const int m    = lane & 15;              // row within the 16-row tile
  const int koff = (lane < 16) ? 0 : 2;    // K-pair per A-layout
  const float* rowp = P + (size_t)(r0 + m) * CTC_C + koff;

  v2f bones = {1.0f, 1.0f};
  v8f acc = {};
#pragma unroll
  for (int kc = 0; kc < 24; ++kc) {
    v2f a = *(const v2f*)(rowp + kc * 4);  // 8B-aligned (row stride 384B)
#if __has_builtin(__builtin_amdgcn_wmma_f32_16x16x4_f32)
    acc = __builtin_amdgcn_wmma_f32_16x16x4_f32(
        false, a, false, bones, (short)0, acc, false, false);
#else
    asm volatile("v_wmma_f32_16x16x4_f32 %0, %1, %2, %0"
                 : "+v"(acc) : "v"(a), "v"(bones));
#endif
  }

  // C/D layout: VGPR r -> lanes 0-15: M=r ; lanes 16-31: M=8+r. N replicated.
  if (lane == 0) {
#pragma unroll
    for (int r = 0; r < 8; ++r)
      nrm[r0 + r] = __logf(acc[r] + CTC_C * EPSV);
  } else if (lane == 16) {
#pragma unroll
    for (int r = 0; r < 8; ++r)
      nrm[r0 + 8 + r] = __logf(acc[r] + CTC_C * EPSV);
  }
}

// ---------------------------------------------------------------------------
// Kernel B: CTC forward recursion, one wave32 per batch element.
// Lane l owns states {l, l+32}; lane 0 additionally owns state 64.
// States l and l+32 have equal parity -> even lanes load only blank prob.
// All cross-state shifts are lane shuffles (no LDS, no barriers).
// ---------------------------------------------------------------------------
__device__ __forceinline__ float lae(float x, float y) {   // logaddexp
  float mx = fmaxf(x, y);
  float mn = fminf(x, y);
  return mx + log1pf(__expf(mn - mx));
}

__global__ void ctc_forward(const int* __restrict__ yt,
                            const float* __restrict__ yp,
                            const float* __restrict__ nrm,
                            float* __restrict__ out) {
  const int lane = threadIdx.x & 31;
  const int b = blockIdx.x * (blockDim.x >> 5) + (threadIdx.x >> 5);

  // Labels distributed one per lane; gather per-state class ids + skip masks.
  const int lab = yt[b * CTC_L + lane];
  const bool odd = (lane & 1) != 0;
  const int j0 = (lane - 1) >> 1;          // state s0 = lane       (odd lanes)
  const int j1 = (lane + 31) >> 1;         // state s1 = lane + 32  (odd lanes)
  const int labj0  = __shfl(lab, j0 & 31);
  const int labj0m = __shfl(lab, (j0 - 1) & 31);
  const int labj1  = __shfl(lab, j1 & 31);
  const int labj1m = __shfl(lab, (j1 - 1) & 31);
  const int c0 = odd ? labj0 : CTC_BLANK;
  const int c1 = odd ? labj1 : CTC_BLANK;
  const bool skip0 = odd && (lane >= 3) && (labj0 != labj0m);
  const bool skip1 = odd && (labj1 != labj1m);   // s1 odd => s1 >= 33 >= 3

  const float* Pb = yp + (size_t)b * CTC_T * CTC_C;
  const float* Nb = nrm + (size_t)b * CTC_T;

  // t = 0: alpha[0] = lp(blank), alpha[1] = lp(label0), rest = NEG.
  float lp_init = __logf(Pb[c0] + EPSV) - Nb[0];
  float a0 = (lane < 2) ? lp_init : NEGV;  // states 0..31
  float a1 = NEGV;                         // states 32..63
  float a2 = NEGV;                         // state 64 (lane 0 only meaningful)

  for (int t = 1; t < CTC_T; ++t) {
    const float* row = Pb + t * CTC_C;
    const float nt = Nb[t];
    float lp0 = __logf(row[c0] + EPSV) - nt;
    float lp1 = odd ? (__logf(row[c1] + EPSV) - nt) : lp0;  // even: same blank

    // Hoist all shuffles of old alpha before any lane-dependent selects.
    float a0u1  = __shfl_up(a0, 1);
    float a1u1  = __shfl_up(a1, 1);
    float a0u2  = __shfl_up(a0, 2);
    float a1u2  = __shfl_up(a1, 2);
    float a0_30 = __shfl(a0, 30);          // alpha[30]
    float a0_31 = __shfl(a0, 31);          // alpha[31]
    float a1_30 = __shfl(a1, 30);          // alpha[62]
    float a1_31 = __shfl(a1, 31);          // alpha[63]

    float p1_0 = (lane == 0) ? NEGV : a0u1;                       // alpha[s0-1]
    float p2_0 = (lane < 2) ? NEGV : a0u2;                        // alpha[s0-2]
    float p1_1 = (lane == 0) ? a0_31 : a1u1;                      // alpha[s1-1]
    float p2_1 = (lane == 0) ? a0_30
               : ((lane == 1) ? a0_31 : a1u2);                    // alpha[s1-2]

    float s0 = lae(a0, p1_0);
    s0 = skip0 ? lae(s0, p2_0) : s0;
    float s1 = lae(a1, p1_1);
    s1 = skip1 ? lae(s1, p2_1) : s1;
    float s2 = lae(a2, a1_31);             // state 64: blank, no skip
    (void)a1_30;

    a0 = s0 + lp0;
    a1 = s1 + lp1;
    a2 = s2 + lp0;                         // lane 0: lp0 is blank lp
  }

  float a63 = __shfl(a1, 31);
  float a64 = __shfl(a2, 0);
  if (lane == 0) out[b] = -lae(a63, a64);
}

extern "C" void kernel_launch(void* const* d_in, const int* in_sizes, int n_in,
                              void* d_out, int out_size, void* d_ws, size_t ws_size,
                              hipStream_t stream) {
  const int*   y_true = (const int*)d_in[0];     // (1024, 32) int32
  const float* y_pred = (const float*)d_in[1];   // (1024, 256, 96) f32
  float* out = (float*)d_out;                    // (1024, 1) f32
  float* nrm = (float*)d_ws;                     // B*T floats = 1 MB scratch

  const int nrows = CTC_B * CTC_T;               // 262144 rows of 96
  // Kernel A: 8 waves/block, 16 rows/wave -> 128 rows/block.
  ctc_rowsum_wmma<<<nrows / 128, 256, 0, stream>>>(y_pred, nrm, nrows);
  // Kernel B: 8 waves/block, 1 batch element per wave.
  ctc_forward<<<CTC_B / 8, 256, 0, stream>>>(y_true, y_pred, nrm, out);
}